// HeterRGCNLayer_13013750907176
// MI455X (gfx1250) — compile-verified
//
#include <hip/hip_runtime.h>
#include <hip/hip_bf16.h>

typedef __attribute__((ext_vector_type(2))) float v2f;
typedef __attribute__((ext_vector_type(8))) float v8f;

#define N_USER 50000
#define N_ITEM 50000
#define NEDGE  800000
#define DIM    128

// ---------------------------------------------------------------------------
// Zero fill
// ---------------------------------------------------------------------------
__global__ void __launch_bounds__(256) zero_f32(float* __restrict__ p, int n) {
  int i = blockIdx.x * 256 + threadIdx.x;
  if (i < n) p[i] = 0.0f;
}

// ---------------------------------------------------------------------------
// GEMM + bias: C[N,128] = A[N,128] @ W[128,128] + b   (fp32, WMMA 16x16x4)
// Block: 256 threads = 8 waves; each wave computes a 16-row stripe (all 128
// cols as 8 column tiles). W + bias staged in LDS (64.5 KB of 320 KB/WGP).
// ---------------------------------------------------------------------------
#if __has_builtin(__builtin_amdgcn_wmma_f32_16x16x4_f32)
#define USE_WMMA_F32 1
#else
#define USE_WMMA_F32 0
#endif

__global__ void __launch_bounds__(256)
gemm_bias_f32(const float* __restrict__ A, const float* __restrict__ W,
              const float* __restrict__ bias, float* __restrict__ C, int N) {
  __shared__ float sW[DIM * DIM];
  __shared__ float sB[DIM];

  const int tid = threadIdx.x;
  // Cooperative load of W (16384 floats) via float4: 16 iters * 256 thr * 4
  for (int i = tid * 4; i < DIM * DIM; i += 256 * 4) {
    *reinterpret_cast<float4*>(sW + i) =
        *reinterpret_cast<const float4*>(W + i);
  }
  if (tid < DIM) sB[tid] = bias[tid];
  __syncthreads();

  const int wave = tid >> 5;
  const int lane = tid & 31;
  const int m    = lane & 15;           // row/col-in-tile for this lane
  const int k0   = (lane >> 4) << 1;    // 0 or 2 : K sub-slot of this lane
  const int row0 = (blockIdx.x * 8 + wave) * 16;

#if USE_WMMA_F32
  // Preload this wave's A fragments: 16 rows x 128 K -> 32 v2f per lane.
  // A-matrix 16x4 f32 layout: lanes 0-15 hold K={0,1}, lanes 16-31 K={2,3}.
  int arow = row0 + m;
  if (arow > N - 1) arow = N - 1;       // clamp; garbage rows masked at store
  const float* aptr = A + (size_t)arow * DIM + k0;
  v2f afrag[32];
#pragma unroll
  for (int kk = 0; kk < 32; ++kk)
    afrag[kk] = *reinterpret_cast<const v2f*>(aptr + kk * 4);

  for (int j = 0; j < 8; ++j) {         // 8 column tiles of 16
    // C/D layout: lane holds column n = j*16 + m for every acc VGPR,
    // so bias init is a per-lane broadcast across the 8 accumulators.
    const float bv = sB[j * 16 + m];
    v8f acc;
#pragma unroll
    for (int r = 0; r < 8; ++r) acc[r] = bv;

#pragma unroll
    for (int kk = 0; kk < 32; ++kk) {
      // B-matrix 4x16 f32: lanes 0-15 rows K={0,1}, lanes 16-31 K={2,3},
      // column n = m, striped across lanes within a VGPR.
      const float* bp = sW + (kk * 4 + k0) * DIM + j * 16 + m;
      v2f bfrag;
      bfrag.x = bp[0];
      bfrag.y = bp[DIM];
      acc = __builtin_amdgcn_wmma_f32_16x16x4_f32(
          /*neg_a=*/false, afrag[kk], /*neg_b=*/false, bfrag,
          /*c_mod=*/(short)0, acc, /*reuse_a=*/false, /*reuse_b=*/false);
    }

    // D layout: VGPR r holds row M = r + (lane/16)*8, column j*16 + m.
#pragma unroll
    for (int r = 0; r < 8; ++r) {
      const int row = row0 + r + ((lane >> 4) << 3);
      if (row < N) C[(size_t)row * DIM + j * 16 + m] = acc[r];
    }
  }
#else
  // Scalar fallback (only if the f32 WMMA builtin is unavailable).
  for (int r = 0; r < 2; ++r) {
    const int row = row0 + (tid & 15) + r * 0; // placeholder structure
  }
  const int row = row0 + (lane & 15) + ((lane >> 4) << 3) * 0;
  for (int rr = wave * 16; rr < wave * 16 + 16; ++rr) {
    const int grow = blockIdx.x * 128 + rr;
    if (grow >= N) continue;
    for (int col = lane; col < DIM; col += 32) {
      float acc = sB[col];
      for (int k = 0; k < DIM; ++k) acc += A[(size_t)grow * DIM + k] * sW[k * DIM + col];
      C[(size_t)grow * DIM + col] = acc;
    }
  }
#endif
}

// ---------------------------------------------------------------------------
// Edge scatter: one wave (32 lanes) per edge; each lane moves 4 floats.
// S[dst] += Wh[src]; cnt[dst] += 1.  Wh and S are L2-resident (25.6 MB each).
// ---------------------------------------------------------------------------
__global__ void __launch_bounds__(256)
scatter_sum_cnt(const float* __restrict__ Wh, const int* __restrict__ src,
                const int* __restrict__ dst, float* __restrict__ S,
                float* __restrict__ cnt, int E) {
  const int t = blockIdx.x * 256 + threadIdx.x;
  const int e = t >> 5;
  const int lane = t & 31;
  if (e >= E) return;
  const int s = src[e];
  const int d = dst[e];
  const float4 v =
      *reinterpret_cast<const float4*>(Wh + (size_t)s * DIM + lane * 4);
  float* out = S + (size_t)d * DIM + lane * 4;
  atomicAdd(out + 0, v.x);
  atomicAdd(out + 1, v.y);
  atomicAdd(out + 2, v.z);
  atomicAdd(out + 3, v.w);
  if (lane == 0) atomicAdd(cnt + d, 1.0f);
}

// ---------------------------------------------------------------------------
// Finalize: user_out = s0/c0 (guarded) + s2/c2 (guarded); s0 in-place in out.
// ---------------------------------------------------------------------------
__global__ void __launch_bounds__(256)
finalize_user(float* __restrict__ out, const float* __restrict__ s2,
              const float* __restrict__ c0, const float* __restrict__ c2,
              int n_elems) {
  const int i = blockIdx.x * 256 + threadIdx.x;
  if (i >= n_elems) return;
  const int node = i >> 7;
  const float a = out[i];
  const float b = s2[i];
  const float cc0 = c0[node];
  const float cc2 = c2[node];
  float r = 0.0f;
  if (cc0 > 0.0f) r += a / cc0;
  if (cc2 > 0.0f) r += b / cc2;
  out[i] = r;
}

__global__ void __launch_bounds__(256)
finalize_item(float* __restrict__ out, const float* __restrict__ c1,
              int n_elems) {
  const int i = blockIdx.x * 256 + threadIdx.x;
  if (i >= n_elems) return;
  const float c = c1[i >> 7];
  out[i] = (c > 0.0f) ? out[i] / c : 0.0f;
}

// ---------------------------------------------------------------------------
// Launch
// ---------------------------------------------------------------------------
extern "C" void kernel_launch(void* const* d_in, const int* in_sizes, int n_in,
                              void* d_out, int out_size, void* d_ws,
                              size_t ws_size, hipStream_t stream) {
  const float* h_user = (const float*)d_in[0];
  const float* h_item = (const float*)d_in[1];
  const int* src0 = (const int*)d_in[2];
  const int* dst0 = (const int*)d_in[3];
  const int* src1 = (const int*)d_in[4];
  const int* dst1 = (const int*)d_in[5];
  const int* src2 = (const int*)d_in[6];
  const int* dst2 = (const int*)d_in[7];
  const float* W0 = (const float*)d_in[8];
  const float* b0 = (const float*)d_in[9];
  const float* W1 = (const float*)d_in[10];
  const float* b1 = (const float*)d_in[11];
  const float* W2 = (const float*)d_in[12];
  const float* b2 = (const float*)d_in[13];

  float* out = (float*)d_out;                 // [2, 50000, 128]
  float* user_out = out;                      // also serves as s0 accumulator
  float* item_out = out + (size_t)N_USER * DIM;  // also s1 accumulator

  // Workspace layout (floats):
  //   Wh   : N_USER*128 (reused for all 3 etypes; N_ITEM == N_USER)
  //   S2   : N_USER*128 (etype-2 sums)
  //   c0,c1,c2 : 50000 each (float counts, contiguous)
  float* Wh = (float*)d_ws;
  float* S2 = Wh + (size_t)N_USER * DIM;
  float* c0 = S2 + (size_t)N_USER * DIM;
  float* c1 = c0 + N_USER;
  float* c2 = c1 + N_ITEM;

  const int out_elems = 2 * N_USER * DIM;          // 12.8M
  const int sum_elems = N_USER * DIM;              // 6.4M
  const int cnt_elems = N_USER + N_ITEM + N_USER;  // 150k

  // 1) zero accumulators (d_out is poisoned by the harness)
  zero_f32<<<(out_elems + 255) / 256, 256, 0, stream>>>(out, out_elems);
  zero_f32<<<(sum_elems + 255) / 256, 256, 0, stream>>>(S2, sum_elems);
  zero_f32<<<(cnt_elems + 255) / 256, 256, 0, stream>>>(c0, cnt_elems);

  const int gemm_blocks_u = (N_USER + 127) / 128;
  const int gemm_blocks_i = (N_ITEM + 127) / 128;
  const int scat_blocks = (NEDGE * 32 + 255) / 256;

  // 2) etype 0: user->user, accumulate sums directly into user_out region
  gemm_bias_f32<<<gemm_blocks_u, 256, 0, stream>>>(h_user, W0, b0, Wh, N_USER);
  scatter_sum_cnt<<<scat_blocks, 256, 0, stream>>>(Wh, src0, dst0, user_out,
                                                   c0, NEDGE);

  // 3) etype 2: item->user, accumulate into S2 (Wh buffer reused)
  gemm_bias_f32<<<gemm_blocks_i, 256, 0, stream>>>(h_item, W2, b2, Wh, N_ITEM);
  scatter_sum_cnt<<<scat_blocks, 256, 0, stream>>>(Wh, src2, dst2, S2, c2,
                                                   NEDGE);

  // 4) etype 1: user->item, accumulate directly into item_out region
  gemm_bias_f32<<<gemm_blocks_u, 256, 0, stream>>>(h_user, W1, b1, Wh, N_USER);
  scatter_sum_cnt<<<scat_blocks, 256, 0, stream>>>(Wh, src1, dst1, item_out,
                                                   c1, NEDGE);

  // 5) per-etype mean + cross-etype sum
  finalize_user<<<(sum_elems + 255) / 256, 256, 0, stream>>>(user_out, S2, c0,
                                                             c2, sum_elems);
  finalize_item<<<(sum_elems + 255) / 256, 256, 0, stream>>>(item_out, c1,
                                                             sum_elems);
}